// CausalSemanticGrouping_18863496364393
// MI455X (gfx1250) — compile-verified
//
#include <hip/hip_runtime.h>
#include <math.h>

typedef __attribute__((ext_vector_type(2))) float v2f;
typedef __attribute__((ext_vector_type(8))) float v8f;

#define BQ 64
#define KS 64
#define NK 1024
#define CC 1088
#define DD 512
#define TEMP_INV 14.285714285714286f
#define EPS_ATTN 1e-7f
#define NORM_EPSF 1e-12f

// ---------------- Kernel 1: normalize slot_embed rows -> slotn ----------------
__global__ __launch_bounds__(256) void k_norm_slots(const float* __restrict__ se,
                                                    float* __restrict__ slotn) {
  const int row = blockIdx.x;       // 0..63
  const int t = threadIdx.x;        // 0..255
  __shared__ float red[256];
  float x0 = se[row * DD + t];
  float x1 = se[row * DD + 256 + t];
  red[t] = x0 * x0 + x1 * x1;
  __syncthreads();
  for (int s = 128; s > 0; s >>= 1) {
    if (t < s) red[t] += red[t + s];
    __syncthreads();
  }
  float sc = 1.0f / fmaxf(sqrtf(red[0]), NORM_EPSF);
  slotn[row * DD + t] = x0 * sc;
  slotn[row * DD + 256 + t] = x1 * sc;
}

// ---------------- Kernel 2: per-key-row inverse L2 norm ----------------
__global__ __launch_bounds__(256) void k_key_invnorm(const float* __restrict__ key,
                                                     float* __restrict__ invn) {
  const int wave = threadIdx.x >> 5, lane = threadIdx.x & 31;
  const int row = blockIdx.x * 8 + wave;            // 0 .. B*N-1
  const float* p = key + (size_t)row * DD;
  float ss = 0.0f;
#pragma unroll
  for (int i = 0; i < 4; ++i) {
    float4 v = *(const float4*)(p + i * 128 + lane * 4);
    ss += v.x * v.x + v.y * v.y + v.z * v.z + v.w * v.w;
  }
#pragma unroll
  for (int off = 16; off > 0; off >>= 1) ss += __shfl_xor(ss, off, 32);
  if (lane == 0) invn[row] = 1.0f / fmaxf(sqrtf(ss), NORM_EPSF);
}

// ---------------- Kernel 3: dots[b,k,c] = slotn[k] . col_n[c] ----------------
// grid: (17 column tiles, 64 batches). Tile 0 = slot columns, tiles 1..16 = keys.
__global__ __launch_bounds__(256) void k_dots(const float* __restrict__ key,
                                              const float* __restrict__ slotn,
                                              const float* __restrict__ invn,
                                              float* __restrict__ dots) {
  const int ct = blockIdx.x;
  const int b = blockIdx.y;
  const int t = threadIdx.x;
  __shared__ float a_lds[64 * 36];
  __shared__ float b_lds[64 * 36];
  const int wave = t >> 5, lane = t & 31, half = lane >> 4, mr = lane & 15;
  const int tm = (wave >> 1) * 16, tnb = (wave & 1) * 32;
  const int lrow = t >> 2, lcb = (t & 3) * 8;
  const float* colsrc =
      (ct == 0) ? slotn : key + ((size_t)b * NK + (size_t)(ct - 1) * 64) * DD;
  v8f acc0 = {}, acc1 = {};

  for (int dk = 0; dk < DD; dk += 32) {
    float4 a0 = *(const float4*)(slotn + (size_t)lrow * DD + dk + lcb);
    float4 a1 = *(const float4*)(slotn + (size_t)lrow * DD + dk + lcb + 4);
    float4 b0 = *(const float4*)(colsrc + (size_t)lrow * DD + dk + lcb);
    float4 b1 = *(const float4*)(colsrc + (size_t)lrow * DD + dk + lcb + 4);
    __syncthreads();  // previous iteration done reading LDS
    *(float4*)&a_lds[lrow * 36 + lcb] = a0;
    *(float4*)&a_lds[lrow * 36 + lcb + 4] = a1;
    *(float4*)&b_lds[lrow * 36 + lcb] = b0;
    *(float4*)&b_lds[lrow * 36 + lcb + 4] = b1;
    __syncthreads();
#pragma unroll
    for (int kk = 0; kk < 32; kk += 4) {
      v2f av = *(const v2f*)&a_lds[(tm + mr) * 36 + kk + 2 * half];
      v2f bv0 = *(const v2f*)&b_lds[(tnb + mr) * 36 + kk + 2 * half];
      v2f bv1 = *(const v2f*)&b_lds[(tnb + 16 + mr) * 36 + kk + 2 * half];
      acc0 = __builtin_amdgcn_wmma_f32_16x16x4_f32(false, av, false, bv0,
                                                   (short)0, acc0, false, false);
      acc1 = __builtin_amdgcn_wmma_f32_16x16x4_f32(false, av, false, bv1,
                                                   (short)0, acc1, false, false);
    }
  }

  {
    int c = ct * 64 + tnb + mr;
    float inv = (ct == 0) ? 1.0f : invn[b * NK + c - KS];
    float* dst = dots + (size_t)b * KS * CC + c;
#pragma unroll
    for (int v = 0; v < 8; ++v) {
      int k = tm + v + 8 * half;
      dst[(size_t)k * CC] = acc0[v] * inv;
    }
  }
  {
    int c = ct * 64 + tnb + 16 + mr;
    float inv = (ct == 0) ? 1.0f : invn[b * NK + c - KS];
    float* dst = dots + (size_t)b * KS * CC + c;
#pragma unroll
    for (int v = 0; v < 8; ++v) {
      int k = tm + v + 8 * half;
      dst[(size_t)k * CC] = acc1[v] * inv;
    }
  }
}

// ---------------- Kernel 4: per-row masked max & inv-sum ----------------
__global__ __launch_bounds__(256) void k_softstats(const float* __restrict__ dots,
                                                   float* __restrict__ rowm,
                                                   float* __restrict__ rowsinv) {
  const int wave = threadIdx.x >> 5, lane = threadIdx.x & 31;
  const int row = blockIdx.x * 8 + wave;  // 0 .. B*K-1
  const int k = row & 63;
  const float* p = dots + (size_t)row * CC;
  float mx = -3.4e38f;
  for (int c = lane; c < CC; c += 32) {
    bool allowed = (c >= KS) || (c < k);
    if (allowed) mx = fmaxf(mx, p[c]);
  }
#pragma unroll
  for (int off = 16; off > 0; off >>= 1) mx = fmaxf(mx, __shfl_xor(mx, off, 32));
  float s = 0.0f;
  for (int c = lane; c < CC; c += 32) {
    bool allowed = (c >= KS) || (c < k);
    if (allowed) s += __expf((p[c] - mx) * TEMP_INV);
  }
#pragma unroll
  for (int off = 16; off > 0; off >>= 1) s += __shfl_xor(s, off, 32);
  if (lane == 0) {
    rowm[row] = mx;
    rowsinv[row] = 1.0f / (s * (1.0f + EPS_ATTN));
  }
}

// ---------------- Kernel 5: out[b] = attn_n[b] @ [slots; key[b]] ----------------
// grid: (8 D-tiles, 64 batches)
__global__ __launch_bounds__(256) void k_out(const float* __restrict__ key,
                                             const float* __restrict__ se,
                                             const float* __restrict__ dots,
                                             const float* __restrict__ rowm,
                                             const float* __restrict__ rowsinv,
                                             float* __restrict__ out) {
  const int dt = blockIdx.x;
  const int b = blockIdx.y;
  const int t = threadIdx.x;
  __shared__ float p_lds[64 * 68];
  __shared__ float v_lds[64 * 68];
  const int wave = t >> 5, lane = t & 31, half = lane >> 4, mr = lane & 15;
  const int tm = (wave >> 1) * 16, tnb = (wave & 1) * 32;
  const int prow = t >> 2, pcb = (t & 3) * 16;
  const float pm = rowm[b * KS + prow];
  const float ps = rowsinv[b * KS + prow];
  v8f acc0 = {}, acc1 = {};

  for (int cb = 0; cb < CC; cb += 64) {
    const float* vsrc =
        (cb == 0) ? (se + (size_t)prow * DD + dt * 64 + pcb)
                  : (key + ((size_t)b * NK + (size_t)(cb - KS) + prow) * DD +
                     dt * 64 + pcb);
    float4 vv[4], px[4];
    const float* drow = dots + ((size_t)b * KS + prow) * CC + cb + pcb;
#pragma unroll
    for (int i = 0; i < 4; ++i) {
      vv[i] = *(const float4*)(vsrc + i * 4);
      px[i] = *(const float4*)(drow + i * 4);
    }
    __syncthreads();  // previous iteration done reading LDS
#pragma unroll
    for (int i = 0; i < 4; ++i) {
      float xs[4] = {px[i].x, px[i].y, px[i].z, px[i].w};
      float vs[4] = {vv[i].x, vv[i].y, vv[i].z, vv[i].w};
#pragma unroll
      for (int j = 0; j < 4; ++j) {
        int c = cb + pcb + i * 4 + j;
        bool allowed = (c >= KS) || (c < prow);
        p_lds[prow * 68 + pcb + i * 4 + j] =
            allowed ? __expf((xs[j] - pm) * TEMP_INV) * ps : 0.0f;
        v_lds[(pcb + i * 4 + j) * 68 + prow] = vs[j];  // transposed store
      }
    }
    __syncthreads();
#pragma unroll
    for (int kk = 0; kk < 64; kk += 4) {
      v2f av = *(const v2f*)&p_lds[(tm + mr) * 68 + kk + 2 * half];
      v2f bv0 = *(const v2f*)&v_lds[(tnb + mr) * 68 + kk + 2 * half];
      v2f bv1 = *(const v2f*)&v_lds[(tnb + 16 + mr) * 68 + kk + 2 * half];
      acc0 = __builtin_amdgcn_wmma_f32_16x16x4_f32(false, av, false, bv0,
                                                   (short)0, acc0, false, false);
      acc1 = __builtin_amdgcn_wmma_f32_16x16x4_f32(false, av, false, bv1,
                                                   (short)0, acc1, false, false);
    }
  }

  {
    int d = dt * 64 + tnb + mr;
    float* dst = out + (size_t)b * KS * DD + d;
#pragma unroll
    for (int v = 0; v < 8; ++v) {
      int k = tm + v + 8 * half;
      dst[(size_t)k * DD] = acc0[v];
    }
  }
  {
    int d = dt * 64 + tnb + 16 + mr;
    float* dst = out + (size_t)b * KS * DD + d;
#pragma unroll
    for (int v = 0; v < 8; ++v) {
      int k = tm + v + 8 * half;
      dst[(size_t)k * DD] = acc1[v];
    }
  }
}

extern "C" void kernel_launch(void* const* d_in, const int* in_sizes, int n_in,
                              void* d_out, int out_size, void* d_ws, size_t ws_size,
                              hipStream_t stream) {
  const float* key = (const float*)d_in[0];   // [64,1024,512]
  const float* se = (const float*)d_in[1];    // [64,512]
  float* out = (float*)d_out;                 // [64,64,512]
  float* dots = out + (size_t)BQ * KS * DD;   // [64,64,1088]

  float* ws = (float*)d_ws;
  float* slotn = ws;                       // 64*512   = 32768 floats
  float* invn = slotn + (size_t)KS * DD;   // 64*1024  = 65536 floats
  float* rowm = invn + (size_t)BQ * NK;    // 64*64    = 4096 floats
  float* rowsv = rowm + (size_t)BQ * KS;   // 64*64    = 4096 floats

  k_norm_slots<<<KS, 256, 0, stream>>>(se, slotn);
  k_key_invnorm<<<(BQ * NK) / 8, 256, 0, stream>>>(key, invn);
  k_dots<<<dim3(CC / 64, BQ), 256, 0, stream>>>(key, slotn, invn, dots);
  k_softstats<<<(BQ * KS) / 8, 256, 0, stream>>>(dots, rowm, rowsv);
  k_out<<<dim3(DD / 64, BQ), 256, 0, stream>>>(key, se, dots, rowm, rowsv, out);
}